// CategoryCrossAttention_32950989095459
// MI455X (gfx1250) — compile-verified
//
#include <hip/hip_runtime.h>
#include <math.h>

// Problem constants (from reference setup_inputs)
#define B_    32
#define T_    2048
#define D_    1024
#define C_    512
#define H_    1024
#define S_    32            // t-slices per batch for the fused flash pass
#define ROWS_ (T_ / S_)     // 64 rows of t per block
#define EPS_  1e-5f

typedef __attribute__((ext_vector_type(2))) float v2f;
typedef __attribute__((ext_vector_type(4))) float v4f;
typedef __attribute__((ext_vector_type(8))) float v8f;

// ---------------------------------------------------------------------------
// K1: LayerNorm of cat_emb -> cn  [B, C]   (one block per batch, 256 thr)
// ---------------------------------------------------------------------------
__global__ void ln_cat_kernel(const float* __restrict__ cat,
                              const float* __restrict__ g,
                              const float* __restrict__ be,
                              float* __restrict__ cn) {
  const int b = blockIdx.x, tid = threadIdx.x;     // 256 threads, 2 elems each
  __shared__ float red[16];
  __shared__ float bc[2];
  const float2 v = ((const float2*)(cat + (size_t)b * C_))[tid];
  float s1 = v.x + v.y;
  float s2 = v.x * v.x + v.y * v.y;
  #pragma unroll
  for (int off = 16; off > 0; off >>= 1) {
    s1 += __shfl_xor(s1, off);
    s2 += __shfl_xor(s2, off);
  }
  const int wid = tid >> 5;
  if ((tid & 31) == 0) { red[wid] = s1; red[8 + wid] = s2; }
  __syncthreads();
  if (tid == 0) {
    float a = 0.f, c2 = 0.f;
    for (int w = 0; w < 8; ++w) { a += red[w]; c2 += red[8 + w]; }
    bc[0] = a; bc[1] = c2;
  }
  __syncthreads();
  const float mu   = bc[0] * (1.0f / C_);
  const float var  = bc[1] * (1.0f / C_) - mu * mu;
  const float rstd = rsqrtf(var + EPS_);
  const int d = 2 * tid;
  cn[(size_t)b * C_ + d]     = (v.x - mu) * rstd * g[d]     + be[d];
  cn[(size_t)b * C_ + d + 1] = (v.y - mu) * rstd * g[d + 1] + be[d + 1];
}

// ---------------------------------------------------------------------------
// K2/K3/K6/K7: small GEMM on the WMMA f32 16x16x4 path.
//   Cm[M,N] = A[M,K] * B[K,N]
//   BT==1: second operand stored as Bt[N,K] (torch Linear weight layout)
// One wave per 16x16 output tile. ISA 7.12.2 f32 layouts:
//   A 16x4 : lane l holds M=(l&15), K = k + 2*(l>>4) + {0,1}
//   B 4x16 : lane l holds N=(l&15), K = k + 2*(l>>4) + {0,1}
//   C/D    : VGPR i holds row M = i + 8*(l>>4), col N=(l&15)
// Templated on BT so the inner loop is branch-free; unrolled 2x (two WMMAs
// per iteration, loads as aligned v2f -> global_load_b64).
// ---------------------------------------------------------------------------
template <int BT>
__global__ void wmma_gemm_f32_kernel(const float* __restrict__ A,
                                     const float* __restrict__ Bm,
                                     float* __restrict__ Cm,
                                     int M, int N, int K) {
  const int gid  = blockIdx.x * blockDim.x + threadIdx.x;
  const int wave = gid >> 5;
  const int lane = threadIdx.x & 31;
  const int tiles_n = N >> 4;
  const int tiles   = (M >> 4) * tiles_n;
  if (wave >= tiles) return;            // whole-wave exit: EXEC stays all-1s
  const int tm = wave / tiles_n;
  const int tn = wave % tiles_n;

  const int mrow  = tm * 16 + (lane & 15);
  const int ncol  = tn * 16 + (lane & 15);
  const int khalf = 2 * (lane >> 4);    // 0 or 2

  const float* ap = A + (size_t)mrow * K + khalf;
  const float* bp = BT ? (Bm + (size_t)ncol * K + khalf)
                       : (Bm + (size_t)khalf * N + ncol);

  v8f c = {};
  for (int k = 0; k < K; k += 8) {
    const v2f a0 = *(const v2f*)(ap);
    const v2f a1 = *(const v2f*)(ap + 4);
    ap += 8;
    v2f b0, b1;
    if (BT) {
      b0 = *(const v2f*)(bp);
      b1 = *(const v2f*)(bp + 4);
      bp += 8;
    } else {
      b0[0] = bp[0];
      b0[1] = bp[N];
      b1[0] = bp[4 * (size_t)N];
      b1[1] = bp[5 * (size_t)N];
      bp += 8 * (size_t)N;
    }
    c = __builtin_amdgcn_wmma_f32_16x16x4_f32(false, a0, false, b0,
                                              (short)0, c, false, false);
    c = __builtin_amdgcn_wmma_f32_16x16x4_f32(false, a1, false, b1,
                                              (short)0, c, false, false);
  }
  const int base_m = tm * 16 + 8 * (lane >> 4);
  #pragma unroll
  for (int i = 0; i < 8; ++i)
    Cm[(size_t)(base_m + i) * N + tn * 16 + (lane & 15)] = c[i];
}

// ---------------------------------------------------------------------------
// K4: fused LayerNorm(x) + logits + online softmax + weighted-x accumulation.
// Dominant pass: streams x (256 MB) exactly once.
// block = (b, s) slice of ROWS_ rows of t; 256 threads, thread owns 4 d-elems.
// Emits per-slice flash partials: m, Z, acc[D].
// ---------------------------------------------------------------------------
__global__ void fused_wei_xbar_kernel(const float* __restrict__ x,
                                      const float* __restrict__ g,
                                      const float* __restrict__ be,
                                      const float* __restrict__ qk,
                                      float* __restrict__ pm,
                                      float* __restrict__ pZ,
                                      float* __restrict__ pacc) {
  const int blk = blockIdx.x;
  const int b   = blk >> 5;            // S_ == 32
  const int s   = blk & (S_ - 1);
  const int t0  = s * ROWS_;
  const int tid = threadIdx.x;         // 256
  const int d0  = tid * 4;
  const int wid = tid >> 5;

  __shared__ float qk_s[D_], g_s[D_], b_s[D_];
  __shared__ float red[16];
  __shared__ float bc[2];

  for (int i = tid; i < D_; i += 256) {
    qk_s[i] = qk[(size_t)b * D_ + i];
    g_s[i]  = g[i];
    b_s[i]  = be[i];
  }
  __syncthreads();

  float m = -1e30f, Z = 0.f;
  float acc0 = 0.f, acc1 = 0.f, acc2 = 0.f, acc3 = 0.f;
  const float scale_h = rsqrtf((float)H_);   // H^-0.5

  const float* rowp = x + ((size_t)b * T_ + t0) * D_ + d0;
  for (int r = 0; r < ROWS_; ++r) {
    const float4 xv = *(const float4*)rowp;
    if (r + 1 < ROWS_)
      __builtin_prefetch(rowp + D_, 0, 1);   // global_prefetch_b8 next row
    rowp += D_;

    // --- block reduce sum / sumsq for LayerNorm stats ---
    float s1 = xv.x + xv.y + xv.z + xv.w;
    float s2 = xv.x * xv.x + xv.y * xv.y + xv.z * xv.z + xv.w * xv.w;
    #pragma unroll
    for (int off = 16; off > 0; off >>= 1) {
      s1 += __shfl_xor(s1, off);
      s2 += __shfl_xor(s2, off);
    }
    if ((tid & 31) == 0) { red[wid] = s1; red[8 + wid] = s2; }
    __syncthreads();
    if (tid == 0) {
      float a = 0.f, c2 = 0.f;
      for (int w = 0; w < 8; ++w) { a += red[w]; c2 += red[8 + w]; }
      bc[0] = a; bc[1] = c2;
    }
    __syncthreads();
    const float mu   = bc[0] * (1.0f / D_);
    const float var  = bc[1] * (1.0f / D_) - mu * mu;
    const float rstd = rsqrtf(var + EPS_);

    const float xn0 = (xv.x - mu) * rstd * g_s[d0]     + b_s[d0];
    const float xn1 = (xv.y - mu) * rstd * g_s[d0 + 1] + b_s[d0 + 1];
    const float xn2 = (xv.z - mu) * rstd * g_s[d0 + 2] + b_s[d0 + 2];
    const float xn3 = (xv.w - mu) * rstd * g_s[d0 + 3] + b_s[d0 + 3];

    // --- block reduce dot(xn, qk) -> logit ---
    float wp = xn0 * qk_s[d0] + xn1 * qk_s[d0 + 1]
             + xn2 * qk_s[d0 + 2] + xn3 * qk_s[d0 + 3];
    #pragma unroll
    for (int off = 16; off > 0; off >>= 1) wp += __shfl_xor(wp, off);
    __syncthreads();                     // all threads done reading bc[0..1]
    if ((tid & 31) == 0) red[wid] = wp;
    __syncthreads();
    if (tid == 0) {
      float a = 0.f;
      for (int w = 0; w < 8; ++w) a += red[w];
      bc[0] = a;
    }
    __syncthreads();
    const float wei = bc[0] * scale_h;

    // --- online softmax update (identical on every thread) ---
    const float m_new = fmaxf(m, wei);
    const float sc    = __expf(m - m_new);
    const float p     = __expf(wei - m_new);
    Z = Z * sc + p;
    acc0 = acc0 * sc + p * xn0;
    acc1 = acc1 * sc + p * xn1;
    acc2 = acc2 * sc + p * xn2;
    acc3 = acc3 * sc + p * xn3;
    m = m_new;
    __syncthreads();                     // bc/red safe for next iteration
  }

  *(float4*)(pacc + (size_t)blk * D_ + d0) = make_float4(acc0, acc1, acc2, acc3);
  if (tid == 0) { pm[blk] = m; pZ[blk] = Z; }
}

// ---------------------------------------------------------------------------
// K5: flash-merge the S_ partials per batch -> xbar[b, d]
// ---------------------------------------------------------------------------
__global__ void combine_kernel(const float* __restrict__ pm,
                               const float* __restrict__ pZ,
                               const float* __restrict__ pacc,
                               float* __restrict__ xbar) {
  const int b = blockIdx.x, tid = threadIdx.x;
  float m = -1e30f;
  for (int s = 0; s < S_; ++s) m = fmaxf(m, pm[b * S_ + s]);
  float Z = 0.f;
  for (int s = 0; s < S_; ++s) Z += pZ[b * S_ + s] * __expf(pm[b * S_ + s] - m);
  const float inv = 1.0f / Z;
  const int d0 = tid * 4;
  float4 acc = make_float4(0.f, 0.f, 0.f, 0.f);
  for (int s = 0; s < S_; ++s) {
    const float w = __expf(pm[b * S_ + s] - m);
    const float4 a = *(const float4*)(pacc + (size_t)(b * S_ + s) * D_ + d0);
    acc.x += w * a.x; acc.y += w * a.y; acc.z += w * a.z; acc.w += w * a.w;
  }
  *(float4*)(xbar + (size_t)b * D_ + d0) =
      make_float4(acc.x * inv, acc.y * inv, acc.z * inv, acc.w * inv);
}

// ---------------------------------------------------------------------------
// K8: residual broadcast add:  y[b,t,:] = x[b,t,:] + proj[b,:]
// 256 MB read + 256 MB write, v4f streaming with non-temporal hints
// (last-use read, write-once output; don't thrash the 192 MB L2).
// NOTE: nontemporal builtins require clang ext_vector types, not float4.
// ---------------------------------------------------------------------------
__global__ void final_add_kernel(const float* __restrict__ x,
                                 const float* __restrict__ proj,
                                 float* __restrict__ y) {
  const size_t i = (size_t)blockIdx.x * blockDim.x + threadIdx.x; // v4f idx
  const int    d4  = (int)(i & 255);          // D_/4 == 256
  const size_t row = i >> 8;                  // b*T + t
  const int    b   = (int)(row >> 11);        // T_ == 2048
  const v4f xv = __builtin_nontemporal_load((const v4f*)x + i);
  const v4f pv = *((const v4f*)(proj + (size_t)b * D_) + d4);
  const v4f o  = xv + pv;
  __builtin_nontemporal_store(o, (v4f*)y + i);
}

// ---------------------------------------------------------------------------
extern "C" void kernel_launch(void* const* d_in, const int* in_sizes, int n_in,
                              void* d_out, int out_size, void* d_ws, size_t ws_size,
                              hipStream_t stream) {
  const float* x    = (const float*)d_in[0];
  const float* cat  = (const float*)d_in[1];
  const float* lnxg = (const float*)d_in[2];
  const float* lnxb = (const float*)d_in[3];
  const float* lncg = (const float*)d_in[4];
  const float* lncb = (const float*)d_in[5];
  const float* Wq   = (const float*)d_in[6];   // [H, C]
  const float* Wk   = (const float*)d_in[7];   // [H, D]
  const float* Wv   = (const float*)d_in[8];   // [H, D]
  const float* Wp   = (const float*)d_in[9];   // [D, H]
  float* y  = (float*)d_out;
  float* ws = (float*)d_ws;

  // workspace layout (floats)
  float* cn   = ws;                         // B*C      = 16384
  float* q    = cn   + (size_t)B_ * C_;     // B*H      = 32768
  float* qk   = q    + (size_t)B_ * H_;     // B*D      = 32768
  float* pm   = qk   + (size_t)B_ * D_;     // B*S      = 1024
  float* pZ   = pm   + (size_t)B_ * S_;     // B*S      = 1024
  float* pacc = pZ   + (size_t)B_ * S_;     // B*S*D    = 1048576
  float* xbar = pacc + (size_t)B_ * S_ * D_;// B*D      = 32768
  float* outv = xbar + (size_t)B_ * D_;     // B*H      = 32768
  float* proj = outv + (size_t)B_ * H_;     // B*D      = 32768

  // K1: cn = LayerNorm(cat_emb)
  ln_cat_kernel<<<B_, 256, 0, stream>>>(cat, lncg, lncb, cn);

  // K2: q = cn @ Wq^T          [32 x 1024], K=512, Bt layout
  {
    const int tiles = (B_ / 16) * (H_ / 16);            // 128 waves
    wmma_gemm_f32_kernel<1><<<(tiles * 32 + 127) / 128, 128, 0, stream>>>(
        cn, Wq, q, B_, H_, C_);
  }
  // K3: qk = q @ Wk            [32 x 1024], K=1024, B natural K x N layout
  {
    const int tiles = (B_ / 16) * (D_ / 16);
    wmma_gemm_f32_kernel<0><<<(tiles * 32 + 127) / 128, 128, 0, stream>>>(
        q, Wk, qk, B_, D_, H_);
  }

  // K4: fused LN + logits + online softmax + weighted accumulation (one x read)
  fused_wei_xbar_kernel<<<B_ * S_, 256, 0, stream>>>(
      x, lnxg, lnxb, qk, pm, pZ, pacc);

  // K5: merge flash partials -> xbar
  combine_kernel<<<B_, 256, 0, stream>>>(pm, pZ, pacc, xbar);

  // K6: out = xbar @ Wv^T      [32 x 1024], K=1024
  {
    const int tiles = (B_ / 16) * (H_ / 16);
    wmma_gemm_f32_kernel<1><<<(tiles * 32 + 127) / 128, 128, 0, stream>>>(
        xbar, Wv, outv, B_, H_, D_);
  }
  // K7: proj = out @ Wp^T      [32 x 1024], K=1024
  {
    const int tiles = (B_ / 16) * (D_ / 16);
    wmma_gemm_f32_kernel<1><<<(tiles * 32 + 127) / 128, 128, 0, stream>>>(
        outv, Wp, proj, B_, D_, H_);
  }

  // K8: y = x + proj[b] broadcast over t  (second x read + output write)
  {
    const size_t n4 = (size_t)B_ * T_ * (D_ / 4);       // 16,777,216 v4f
    final_add_kernel<<<(unsigned)(n4 / 256), 256, 0, stream>>>(x, proj, y);
  }
}